// BilinearMixture_51041391346342
// MI455X (gfx1250) — compile-verified
//
#include <hip/hip_runtime.h>

typedef __attribute__((ext_vector_type(2))) float v2f;
typedef __attribute__((ext_vector_type(8))) float v8f;

#define D_DIM 64
#define K_BASIS 3
#define C_CLS 5
#define C_PAD 16   // pad classes to 16 for the WMMA N dimension

// ---------------------------------------------------------------------------
// Kernel 1: fuse W (3x64) and scalars (3x5) into padded Wc (16x64):
//   Wc[n][d] = sum_k scalars[k][n] * W[k][d]  for n < 5, else 0.
// ---------------------------------------------------------------------------
__global__ void build_wc_kernel(const float* __restrict__ W,
                                const float* __restrict__ scalars,
                                float* __restrict__ WcB) {
    int t = blockIdx.x * blockDim.x + threadIdx.x;
    if (t >= C_PAD * D_DIM) return;
    int n = t / D_DIM;
    int d = t % D_DIM;
    float acc = 0.0f;
    if (n < C_CLS) {
        #pragma unroll
        for (int k = 0; k < K_BASIS; ++k)
            acc += scalars[k * C_CLS + n] * W[k * D_DIM + d];
    }
    WcB[t] = acc;
}

// ---------------------------------------------------------------------------
// Kernel 2: main edge kernel. One wave32 processes a tile of 16 edges:
//   P[16x64] = u_g * v_g (elementwise, built on the fly per K=4 chunk)
//   Dtile[16x16] = P @ Wc^T  via 16 chained V_WMMA_F32_16X16X4_F32
// A-frag layout (16x4 f32): lane (n + 16h) holds {P[n, 4kc+2h], P[n, 4kc+2h+1]}
// B-frag layout (4x16 f32): lane (n + 16h) holds {Wc[n][4kc+2h], Wc[n][4kc+2h+1]}
// D layout (16x16 f32):     lane (n + 16h), VGPR j holds D[8h+j][n]
// ---------------------------------------------------------------------------
__global__ void bilinear_edges_kernel(const float* __restrict__ u_feat,
                                      const float* __restrict__ v_feat,
                                      const float* __restrict__ WcB,
                                      const float* __restrict__ u_bias,
                                      const float* __restrict__ v_bias,
                                      const int*   __restrict__ u_idx,
                                      const int*   __restrict__ v_idx,
                                      float*       __restrict__ out,
                                      int numTiles) {
    const int lane  = threadIdx.x & 31;
    const int waveL = threadIdx.x >> 5;
    const int wavesPerBlock = blockDim.x >> 5;
    const int gwave  = blockIdx.x * wavesPerBlock + waveL;
    const int nwaves = gridDim.x * wavesPerBlock;

    const int n = lane & 15;   // row role (A) / column role (B, D)
    const int h = lane >> 4;   // which half of the K=4 / M=16 split

    // Preload all 16 B fragments (Wc is 4 KB, L2/L0 resident; registers after)
    v2f bfrag[16];
    #pragma unroll
    for (int kc = 0; kc < 16; ++kc) {
        const float* p = WcB + n * D_DIM + kc * 4 + 2 * h;
        bfrag[kc] = *(const v2f*)p;
    }

    for (int tile = gwave; tile < numTiles; tile += nwaves) {
        const int e  = tile * 16 + n;           // this lane's A-matrix row (edge)
        const int ui = u_idx[e];
        const int vi = v_idx[e];
        const float* up = u_feat + (size_t)ui * D_DIM + 2 * h;
        const float* vp = v_feat + (size_t)vi * D_DIM + 2 * h;

        v8f acc = {};
        #pragma unroll
        for (int kc = 0; kc < 16; ++kc) {
            v2f uu = *(const v2f*)(up + kc * 4);
            v2f vv = *(const v2f*)(vp + kc * 4);
            v2f a;
            a.x = uu.x * vv.x;
            a.y = uu.y * vv.y;
            // 8 args: (neg_a, A, neg_b, B, c_mod, C, reuse_a, reuse_b)
            acc = __builtin_amdgcn_wmma_f32_16x16x4_f32(
                false, a, false, bfrag[kc], (short)0, acc, false, false);
        }

        // Store valid classes (cols 0..4) + gathered biases.
        if (n < C_CLS) {
            #pragma unroll
            for (int j = 0; j < 8; ++j) {
                const int m  = 8 * h + j;       // edge-in-tile for this D element
                const int e2 = tile * 16 + m;
                const int ui2 = u_idx[e2];
                const int vi2 = v_idx[e2];
                const float val = acc[j]
                                + u_bias[(size_t)ui2 * C_CLS + n]
                                + v_bias[(size_t)vi2 * C_CLS + n];
                out[(size_t)e2 * C_CLS + n] = val;
            }
        }
    }
}

// ---------------------------------------------------------------------------
// Kernel 3: scalar tail for E % 16 edges (not hit for E = 2,000,000).
// ---------------------------------------------------------------------------
__global__ void bilinear_tail_kernel(const float* __restrict__ u_feat,
                                     const float* __restrict__ v_feat,
                                     const float* __restrict__ WcB,
                                     const float* __restrict__ u_bias,
                                     const float* __restrict__ v_bias,
                                     const int*   __restrict__ u_idx,
                                     const int*   __restrict__ v_idx,
                                     float*       __restrict__ out,
                                     int tailBase, int tailCount) {
    int t = blockIdx.x * blockDim.x + threadIdx.x;
    if (t >= tailCount) return;
    int e = tailBase + t;
    const int ui = u_idx[e];
    const int vi = v_idx[e];
    const float* up = u_feat + (size_t)ui * D_DIM;
    const float* vp = v_feat + (size_t)vi * D_DIM;
    float accs[C_CLS];
    #pragma unroll
    for (int c = 0; c < C_CLS; ++c) accs[c] = 0.0f;
    for (int d = 0; d < D_DIM; ++d) {
        float p = up[d] * vp[d];
        #pragma unroll
        for (int c = 0; c < C_CLS; ++c)
            accs[c] += p * WcB[c * D_DIM + d];
    }
    #pragma unroll
    for (int c = 0; c < C_CLS; ++c) {
        out[(size_t)e * C_CLS + c] = accs[c]
            + u_bias[(size_t)ui * C_CLS + c]
            + v_bias[(size_t)vi * C_CLS + c];
    }
}

extern "C" void kernel_launch(void* const* d_in, const int* in_sizes, int n_in,
                              void* d_out, int out_size, void* d_ws, size_t ws_size,
                              hipStream_t stream) {
    const float* u_feat  = (const float*)d_in[0];
    const float* v_feat  = (const float*)d_in[1];
    const float* W       = (const float*)d_in[2];
    const float* scalars = (const float*)d_in[3];
    const float* u_bias  = (const float*)d_in[4];
    const float* v_bias  = (const float*)d_in[5];
    const int*   u_idx   = (const int*)d_in[6];
    const int*   v_idx   = (const int*)d_in[7];
    float* out = (float*)d_out;

    const int E = in_sizes[6];

    float* WcB = (float*)d_ws;  // 16*64*4 = 4 KB of workspace

    // Build fused, class-padded Wc.
    {
        int total = C_PAD * D_DIM;
        int threads = 256;
        int blocks = (total + threads - 1) / threads;
        build_wc_kernel<<<blocks, threads, 0, stream>>>(W, scalars, WcB);
    }

    const int numTiles = E / 16;
    const int tail     = E % 16;

    if (numTiles > 0) {
        const int threads = 256;                 // 8 waves per block
        const int wavesPerBlock = threads / 32;
        int blocks = (numTiles + wavesPerBlock - 1) / wavesPerBlock;
        if (blocks > 4096) blocks = 4096;        // grid-stride over tiles
        bilinear_edges_kernel<<<blocks, threads, 0, stream>>>(
            u_feat, v_feat, WcB, u_bias, v_bias, u_idx, v_idx, out, numTiles);
    }

    if (tail > 0) {
        bilinear_tail_kernel<<<1, 64, 0, stream>>>(
            u_feat, v_feat, WcB, u_bias, v_bias, u_idx, v_idx, out,
            numTiles * 16, tail);
    }
}